// AuxiliaryYFixed_6322191860465
// MI455X (gfx1250) — compile-verified
//
#include <hip/hip_runtime.h>
#include <math.h>

typedef __attribute__((ext_vector_type(2))) float v2f;
typedef __attribute__((ext_vector_type(8))) float v8f;

// ---------------------------------------------------------------------------
// Kernel 1: init per-bag max buffer to -inf
// ---------------------------------------------------------------------------
__global__ void init_max_kernel(float* __restrict__ M, int B) {
    int i = blockIdx.x * blockDim.x + threadIdx.x;
    if (i < B) M[i] = -INFINITY;
}

// ---------------------------------------------------------------------------
// Kernel 2: fused GEMV (via v_wmma_f32_16x16x4_f32) + sorted segment-max.
// One wave handles a 16-row tile per iteration.
//
// WMMA mapping (D = A x B + C, 16x16 = 16x4 * 4x16, chained over K=128):
//   A[m][k] = W[k]            (replicated over m; built from scalar loads)
//   B[k][n] = z[row0+n][k]    (native layout == one float2 global load/lane)
//   => D[m][n] = dot(z[row0+n], W)  for every m; take accumulator VGPR0.
// ---------------------------------------------------------------------------
__global__ __launch_bounds__(256) void gemv_segmax_kernel(
        const float* __restrict__ z,    // [N,128]
        const int*   __restrict__ bag,  // [N], sorted ascending
        const float* __restrict__ W,    // [128]
        const float* __restrict__ bvec, // [1]
        float* __restrict__ M,          // [B]  (pre-inited to -inf)
        float* __restrict__ loc,        // [N]
        int N, int numTiles)
{
    const int lane = threadIdx.x & 31;
    const int wave = threadIdx.x >> 5;
    const int wavesPerBlock = blockDim.x >> 5;
    const int h = lane >> 4;   // half-wave (selects K pair: +0 or +2)
    const int n = lane & 15;   // column of the tile (row within 16-row group)
    const float bias = bvec[0];

    for (int tile = blockIdx.x * wavesPerBlock + wave; tile < numTiles;
         tile += gridDim.x * wavesPerBlock) {
        const int row0 = tile * 16;
        const int row  = row0 + n;
        const int rowc = (row < N) ? row : (N - 1);   // clamp (keeps EXEC full)
        const float* zr = z + (long)rowc * 128 + 2 * h;

        v8f c = {};
#pragma unroll
        for (int k0 = 0; k0 < 128; k0 += 4) {
            // B-matrix fragment: lane(h,n), vgpr v -> z[rowc][k0 + 2h + v]
            v2f bm = *(const v2f*)(zr + k0);
            // A-matrix fragment: uniform W chunk -> s_load, 2 cndmasks
            float4 wk = *(const float4*)(W + k0);
            v2f a;
            a[0] = h ? wk.z : wk.x;   // K = k0 + 2h + 0
            a[1] = h ? wk.w : wk.y;   // K = k0 + 2h + 1
            c = __builtin_amdgcn_wmma_f32_16x16x4_f32(
                    /*neg_a=*/false, a, /*neg_b=*/false, bm,
                    /*c_mod=*/(short)0, c, /*reuse_a=*/false, /*reuse_b=*/false);
        }

        // Accumulator VGPR0: lanes 0-15 -> D[0][n], lanes 16-31 -> D[8][n];
        // both equal dot(z[row0+n], W). Every lane holds loc[row0 + n].
        const float locv = c[0] + bias;
        const int bg = bag[rowc];

        // Store loc (lanes 0-15 carry n = 0..15 exactly once).
        if (lane < 16 && row < N) loc[row] = locv;

        // Segmented inclusive max-scan over n = 0..15 (mirrored in each half;
        // the n >= d guard keeps pulls within a half-wave).
        float mval = locv;
#pragma unroll
        for (int d = 1; d < 16; d <<= 1) {
            float oV = __shfl_up(mval, (unsigned)d, 32);
            int   oB = __shfl_up(bg,   (unsigned)d, 32);
            if (n >= d && oB == bg) mval = fmaxf(mval, oV);
        }
        // Last lane of each in-tile bag run flushes one atomic.
        const int nb = __shfl_down(bg, 1u, 32);
        const bool isLast = (n == 15) || (nb != bg);
        if (lane < 16 && row < N && isLast) {
            atomicMax(&M[bg], mval);
        }
    }
}

// ---------------------------------------------------------------------------
// Kernel 3: empty bags (still -inf) -> 0.0, matching the reference's
// counts>0 masking (all real logits are finite).
// ---------------------------------------------------------------------------
__global__ void finalize_kernel(float* __restrict__ M, int B) {
    int i = blockIdx.x * blockDim.x + threadIdx.x;
    if (i < B) {
        float v = M[i];
        if (v == -INFINITY) M[i] = 0.0f;
    }
}

// ---------------------------------------------------------------------------
// Inputs (setup_inputs order): z_ins [N*128] f32, bag_idx [N] i32,
//                              W [128] f32, b [1] f32
// Output: d_out = [ M (B floats) | loc (N floats) ]
// ---------------------------------------------------------------------------
extern "C" void kernel_launch(void* const* d_in, const int* in_sizes, int n_in,
                              void* d_out, int out_size, void* d_ws, size_t ws_size,
                              hipStream_t stream) {
    const float* z   = (const float*)d_in[0];
    const int*   bag = (const int*)  d_in[1];
    const float* W   = (const float*)d_in[2];
    const float* b   = (const float*)d_in[3];

    const int N  = in_sizes[1];          // number of instances
    const int B  = out_size - N;         // number of bags (16384)
    float* M   = (float*)d_out;
    float* loc = M + B;

    const int numTiles = (N + 15) / 16;

    init_max_kernel<<<(B + 255) / 256, 256, 0, stream>>>(M, B);

    int blocks = (numTiles + 7) / 8;     // 8 waves per 256-thread block
    if (blocks > 4096) blocks = 4096;    // grid-stride persistent waves
    gemv_segmax_kernel<<<blocks, 256, 0, stream>>>(z, bag, W, b, M, loc,
                                                   N, numTiles);

    finalize_kernel<<<(B + 255) / 256, 256, 0, stream>>>(M, B);
}